// PostProcessHOI_32856499814727
// MI455X (gfx1250) — compile-verified
//
#include <hip/hip_runtime.h>
#include <hip/hip_bf16.h>
#include <cstdint>

// Problem geometry (fixed by the reference).
#define B_DIM 128
#define N_DIM 2048
#define R_DIM 117
// Flat float32 output offsets (tuple return order: boxes, verb_scores, labels, keep)
#define BOX_OFF   0LL                                       // [B, 2N, 4]
#define VERB_OFF  (BOX_OFF + (long long)B_DIM*2*N_DIM*4)    // [B, N, R]
#define LAB_OFF   (VERB_OFF + (long long)B_DIM*N_DIM*R_DIM) // [B, 2N]
#define KEEP_OFF  (LAB_OFF + (long long)B_DIM*2*N_DIM)      // [B, N]

#define ROWS_PER_WAVE 4
#define WAVES_PER_BLK 8
#define BLOCK_THREADS (WAVES_PER_BLK * 32)
#define DW_PER_WAVE   (ROWS_PER_WAVE * R_DIM)      // 468 dwords = 1872 B (16B aligned)
#define XF_PER_WAVE   (DW_PER_WAVE / 4)            // 117 lane-transfers of 16 B
#define XF_FULL       (XF_PER_WAVE / 32)           // 3 full b128 async issues
#define XF_TAIL       (XF_PER_WAVE - XF_FULL * 32) // 21-lane tail issue

__global__ __launch_bounds__(BLOCK_THREADS)
void PostProcessHOI_kernel(const float* __restrict__ rel,      // [B,N,R]
                           const float* __restrict__ sub_box,  // [B,N,4]
                           const float* __restrict__ obj_box,  // [B,N,4]
                           const int*   __restrict__ sub_cat,  // [B,N]
                           const int*   __restrict__ obj_cat,  // [B,N]
                           const float* __restrict__ orig_sz,  // [B,2] (H,W)
                           const float* __restrict__ cur_sz,   // [B,2] (H,W)
                           float* __restrict__ out)
{
    __shared__ __attribute__((aligned(16))) float smem[WAVES_PER_BLK * DW_PER_WAVE];

    const int lane = threadIdx.x & 31;
    const int wave = threadIdx.x >> 5;
    const long long rowBase =
        (long long)blockIdx.x * (WAVES_PER_BLK * ROWS_PER_WAVE) + (long long)wave * ROWS_PER_WAVE;

    // Wave-uniform batch index: 4 consecutive rows always lie in one batch (N % 4 == 0).
    const int b  = (int)(rowBase >> 11);             // N = 2048
    const int n0 = (int)(rowBase & (N_DIM - 1));
    // sizes are (H, W): scale vector = [sw, sh, sw, sh]   (scalar-friendly, hoisted)
    const float sw = orig_sz[2 * b + 1] / cur_sz[2 * b + 1];
    const float sh = orig_sz[2 * b + 0] / cur_sz[2 * b + 0];

    // ---------------- async global -> LDS staging (b128: 512 B per issue) ----------------
    // GVS mode: mem = SADDR(sgpr pair) + VADDR(32b lane offset); LDS byte addr in first v-op.
    // Zero VGPR destinations: all 4 requests stay in flight, drained by one s_wait_asynccnt.
    const float* gsrc = rel + rowBase * R_DIM;
    float* lbase = &smem[wave * DW_PER_WAVE];
    const uint32_t lds0 = (uint32_t)(uintptr_t)lbase;  // low 32 bits of flat == LDS byte addr

#pragma unroll
    for (int i = 0; i < XF_FULL; ++i) {
        const uint32_t voff  = (uint32_t)((i * 32 + lane) * 16);
        const uint32_t laddr = lds0 + voff;
        asm volatile("global_load_async_to_lds_b128 %0, %1, %2"
                     :: "v"(laddr), "v"(voff), "s"(gsrc) : "memory");
    }
    if (lane < XF_TAIL) {
        const uint32_t voff  = (uint32_t)((XF_FULL * 32 + lane) * 16);
        const uint32_t laddr = lds0 + voff;
        asm volatile("global_load_async_to_lds_b128 %0, %1, %2"
                     :: "v"(laddr), "v"(voff), "s"(gsrc) : "memory");
    }
#if __has_builtin(__builtin_amdgcn_s_wait_asynccnt)
    __builtin_amdgcn_s_wait_asynccnt(0);
#else
    asm volatile("s_wait_asynccnt 0" ::: "memory");
#endif
    // Each wave reads only the LDS it filled itself -> no workgroup barrier needed.

    // ---------------- per-row softmax-threshold; verb scores written back to LDS ----------------
    for (int rr = 0; rr < ROWS_PER_WAVE; ++rr) {
        const int n = n0 + rr;
        float* lrow = lbase + rr * R_DIM;

        // 117 = 3*32 + 21 elements, striped over lanes (bank-conflict free).
        const float s0 = lrow[lane];
        const float s1 = lrow[lane + 32];
        const float s2 = lrow[lane + 64];
        const bool  tail = (lane < (R_DIM - 96));    // 21 lanes
        const float s3 = tail ? lrow[lane + 96] : -__builtin_inff();

        // wave32 max reduction
        float m = fmaxf(fmaxf(s0, s1), fmaxf(s2, s3));
#pragma unroll
        for (int off = 16; off >= 1; off >>= 1)
            m = fmaxf(m, __shfl_xor(m, off, 32));

        // max softmax prob = 1 / sum(exp(s - max));  keep <=> sum <= 2
        float e = __expf(s0 - m) + __expf(s1 - m) + __expf(s2 - m)
                + (tail ? __expf(s3 - m) : 0.0f);
#pragma unroll
        for (int off = 16; off >= 1; off >>= 1)
            e += __shfl_xor(e, off, 32);

        const bool  keep = (e <= 2.0f);
        const float kf   = keep ? 1.0f : 0.0f;

        // verb_scores = raw * keep : write back in place into LDS for the async store below.
        lrow[lane]      = s0 * kf;
        lrow[lane + 32] = s1 * kf;
        lrow[lane + 64] = s2 * kf;
        if (tail) lrow[lane + 96] = s3 * kf;

        // boxes (float4 lanes 0..1) / labels (lanes 2..3) / keep (lane 4)
        if (lane < 2) {
            const float4* src = (const float4*)((lane == 0) ? sub_box : obj_box);
            const float4 bx = src[(long long)b * N_DIM + n];
            float4 r;
            r.x = bx.x * sw * kf;
            r.y = bx.y * sh * kf;
            r.z = bx.z * sw * kf;
            r.w = bx.w * sh * kf;
            ((float4*)(out + BOX_OFF))[(long long)b * (2 * N_DIM) + ((lane == 0) ? n : (N_DIM + n))] = r;
        } else if (lane == 2) {
            out[LAB_OFF + (long long)b * (2 * N_DIM) + n] =
                keep ? (float)(sub_cat[(long long)b * N_DIM + n] - 1) : -1.0f;
        } else if (lane == 3) {
            out[LAB_OFF + (long long)b * (2 * N_DIM) + N_DIM + n] =
                keep ? (float)(obj_cat[(long long)b * N_DIM + n] - 1) : -1.0f;
        } else if (lane == 4) {
            out[KEEP_OFF + rowBase + rr] = kf;
        }
    }

    // ---------------- async LDS -> global streaming of verb_scores (b128) ----------------
    // Drain this wave's in-order DS stores first so the async engine reads final data.
    asm volatile("s_wait_dscnt 0" ::: "memory");
    const float* gdst = out + VERB_OFF + rowBase * R_DIM;   // 16B-aligned, 1872 B contiguous
#pragma unroll
    for (int i = 0; i < XF_FULL; ++i) {
        const uint32_t voff  = (uint32_t)((i * 32 + lane) * 16);
        const uint32_t laddr = lds0 + voff;
        asm volatile("global_store_async_from_lds_b128 %0, %1, %2"
                     :: "v"(voff), "v"(laddr), "s"(gdst) : "memory");
    }
    if (lane < XF_TAIL) {
        const uint32_t voff  = (uint32_t)((XF_FULL * 32 + lane) * 16);
        const uint32_t laddr = lds0 + voff;
        asm volatile("global_store_async_from_lds_b128 %0, %1, %2"
                     :: "v"(voff), "v"(laddr), "s"(gdst) : "memory");
    }
    // S_ENDPGM performs an implicit wait-idle -> outstanding async stores complete before exit.
}

extern "C" void kernel_launch(void* const* d_in, const int* in_sizes, int n_in,
                              void* d_out, int out_size, void* d_ws, size_t ws_size,
                              hipStream_t stream) {
    (void)in_sizes; (void)n_in; (void)out_size; (void)d_ws; (void)ws_size;
    const float* rel     = (const float*)d_in[0];
    const float* sub_box = (const float*)d_in[1];
    const float* obj_box = (const float*)d_in[2];
    const int*   sub_cat = (const int*)  d_in[3];
    const int*   obj_cat = (const int*)  d_in[4];
    const float* orig_sz = (const float*)d_in[5];
    const float* cur_sz  = (const float*)d_in[6];
    float* out = (float*)d_out;

    const long long rows = (long long)B_DIM * N_DIM;                  // 262,144
    const int blocks = (int)(rows / (WAVES_PER_BLK * ROWS_PER_WAVE)); // 8,192

    PostProcessHOI_kernel<<<blocks, BLOCK_THREADS, 0, stream>>>(
        rel, sub_box, obj_box, sub_cat, obj_cat, orig_sz, cur_sz, out);
}